// bilateralFilter_65798898974996
// MI455X (gfx1250) — compile-verified
//
#include <hip/hip_runtime.h>

// ---------------- problem constants (from the reference) ----------------
constexpr int   Bdim = 16;
constexpr int   Hdim = 768;
constexpr int   Wdim = 1280;
constexpr int   PAD  = 4;            // KERNEL_SIZE/2
constexpr int   WV   = Wdim / 4;     // 320 float4-vectors per row
constexpr float EPSF = 1e-12f;

// w = exp(-16/9) * exp(-diff^2/8) = exp2(C0 + C1*diff^2)
constexpr float C0 = -2.5647911838025352f;   // -16/9 * log2(e)
constexpr float C1 = -0.18033688011112043f;  // -log2(e)/8

typedef float v4f __attribute__((ext_vector_type(4)));

__device__ __forceinline__ float wexp(float d) {
    return __builtin_amdgcn_exp2f(fmaf(d * d, C1, C0));   // single v_exp_f32
}

// One block == one image row: 320 lanes of float4, 10 wave32 waves.
__global__ __launch_bounds__(WV) void bilateral_collapsed_kernel(
    const float* __restrict__ left,
    const float* __restrict__ est,
    float* __restrict__ out)
{
    __shared__ v4f sdp[WV];          // async-staged estDisp row segment (5 KB)

    const int x4 = threadIdx.x;      // vector column
    const int y  = blockIdx.y;
    const int b  = blockIdx.z;

    const long long base = ((long long)(b * Hdim + y)) * Wdim + (long long)x4 * 4;

    // Shifts are exactly +-4 floats = +-16B -> all accesses stay b128-aligned,
    // and validity is per-vector: only x4==0 / x4==WV-1 columns are edge cases.
    const bool dp_ok = (y >= PAD) && (x4 >= 1);          // est(y-4, x-4) vector in-range
    const bool ls_ok = (y + PAD < Hdim) && (x4 + 1 < WV);// left(y+4, x+4) vector in-range

    // ---- Stage the read-once est stream via the CDNA5 async copy engine ----
    // Issued first so the global->LDS DMA overlaps the left loads + exp math.
    // Wave-local (each lane fills its own 16B slot) -> no workgroup barrier,
    // only ASYNCcnt ordering.
    if (dp_ok) {
        unsigned long long ga =
            (unsigned long long)(est + base - (long long)PAD * Wdim - PAD);
        unsigned la = (unsigned)(unsigned long long)&sdp[x4]; // LDS byte offset
        asm volatile("global_load_async_to_lds_b128 %0, %1, off"
                     :: "v"(la), "v"(ga) : "memory");
    }

    // ---- left stream: center + shifted reads (regular temporal, L2 reuse) ----
    v4f lc = *(const v4f*)(left + base);
    v4f ls = (v4f){0.f, 0.f, 0.f, 0.f};
    if (ls_ok) ls = *(const v4f*)(left + base + (long long)PAD * Wdim + PAD);

    // ---- range weight while the async copy is in flight ----
    v4f diff = lc - ls;
    v4f w;
    w.x = wexp(diff.x);
    w.y = wexp(diff.y);
    w.z = wexp(diff.z);
    w.w = wexp(diff.w);

    // ---- consume staged est ----
    asm volatile("s_wait_asynccnt 0x0" ::: "memory");
    v4f dp = (v4f){0.f, 0.f, 0.f, 0.f};
    if (dp_ok) dp = sdp[x4];                              // ds_load_b128

    v4f r;
    r.x = fmaf(dp.x, w.x, EPSF) / (w.x + EPSF);
    r.y = fmaf(dp.y, w.y, EPSF) / (w.y + EPSF);
    r.z = fmaf(dp.z, w.z, EPSF) / (w.z + EPSF);
    r.w = fmaf(dp.w, w.w, EPSF) / (w.w + EPSF);

    // Write-once output -> nontemporal store keeps L2 for the left stream.
    __builtin_nontemporal_store(r, (v4f*)(out + base));
}

extern "C" void kernel_launch(void* const* d_in, const int* in_sizes, int n_in,
                              void* d_out, int out_size, void* d_ws, size_t ws_size,
                              hipStream_t stream) {
    (void)in_sizes; (void)n_in; (void)d_ws; (void)ws_size; (void)out_size;
    const float* left = (const float*)d_in[0];   // leftImage, f32 [16,1,768,1280]
    const float* est  = (const float*)d_in[1];   // estDisp,   f32 [16,1,768,1280]
    float*       out  = (float*)d_out;

    dim3 block(WV, 1, 1);            // 320 threads = 10 wave32 waves, one row
    dim3 grid(1, Hdim, Bdim);        // 768 x 16 rows
    bilateral_collapsed_kernel<<<grid, block, 0, stream>>>(left, est, out);
}